// NLinearMemoryEfficient_86474871538346
// MI455X (gfx1250) — compile-verified
//
#include <hip/hip_runtime.h>

typedef __attribute__((ext_vector_type(2))) float v2f;
typedef __attribute__((ext_vector_type(8))) float v8f;

#define B_ROWS   8192
#define N_FEAT   256
#define D_IN     64
#define D_OUT    128
#define BM       128              // rows of B handled per workgroup (8 waves x 16)
#define W_STRIDE 68               // padded k-stride (floats) for LDS W tile: avoids 16-way bank conflicts

__global__ __launch_bounds__(256)
void nlinear_wmma_f32_kernel(const float* __restrict__ x,
                             const float* __restrict__ W,
                             const float* __restrict__ bias,
                             float* __restrict__ out)
{
    __shared__ float w_lds[D_OUT * W_STRIDE];   // w_lds[o * 68 + i] = W[n][o][i]
    __shared__ float b_lds[D_OUT];

    const int n        = blockIdx.x;            // feature index 0..255
    const int rowBlock = blockIdx.y;            // 0..63
    const int tid      = threadIdx.x;           // 0..255
    const int wave     = tid >> 5;
    const int lane     = tid & 31;

    // ---- Stage W[n] (128x64 f32 = 32KB) and bias into LDS ------------------
    // 8192 floats = 2048 float4 loads; 256 threads x 8 iterations.
    const float* Wn = W + (size_t)n * (D_OUT * D_IN);
    #pragma unroll
    for (int it = 0; it < 8; ++it) {
        int idx4 = it * 256 + tid;              // float4 index 0..2047
        int o    = idx4 >> 4;                   // 16 float4 per d_out row
        int i4   = idx4 & 15;
        float4 v = ((const float4*)Wn)[idx4];
        float* dst = &w_lds[o * W_STRIDE + i4 * 4];   // 16B aligned (68*4 % 16 == 0)
        *(float4*)dst = v;
    }
    if (tid < D_OUT) b_lds[tid] = bias[(size_t)n * D_OUT + tid];
    __syncthreads();

    // ---- Load this wave's 16x64 A panel into registers ---------------------
    // WMMA f32 16x16x4 A layout: lane = (k>=2)*16 + m, vgpr = k&1.
    // For fragment kk (k0 = 4*kk): lane holds the contiguous pair
    //   x[row = r0 + (lane&15)][k0 + 2*(lane>>4) + {0,1}]  -> one b64 load.
    const int r0 = rowBlock * BM + wave * 16;
    const int m_lane = lane & 15;
    const int khalf  = lane >> 4;               // 0 or 1

    const float* xrow = x + ((size_t)(r0 + m_lane) * N_FEAT + n) * D_IN + 2 * khalf;
    v2f afrag[16];
    #pragma unroll
    for (int kk = 0; kk < 16; ++kk) {
        afrag[kk] = *(const v2f*)(xrow + kk * 4);
    }

    // ---- 8 output column tiles of 16 cols each -----------------------------
    const size_t row_stride = (size_t)N_FEAT * D_OUT;   // floats between consecutive b rows
    #pragma unroll
    for (int j = 0; j < 8; ++j) {
        v8f acc = {};
        // B layout (4x16, K x N): lane = (k>=2)*16 + col, vgpr = k&1.
        // Lane reads w_lds[(col)*68 + k0 + 2*khalf + {0,1}] -> ds_load_b64,
        // 16 lanes -> 16 distinct banks thanks to the 68-float stride.
        const float* wb = &w_lds[(j * 16 + m_lane) * W_STRIDE + 2 * khalf];
        #pragma unroll
        for (int kk = 0; kk < 16; ++kk) {
            v2f bfrag = *(const v2f*)(wb + kk * 4);
            acc = __builtin_amdgcn_wmma_f32_16x16x4_f32(
                      /*neg_a=*/false, afrag[kk],
                      /*neg_b=*/false, bfrag,
                      /*c_mod=*/(short)0, acc,
                      /*reuse_a=*/false, /*reuse_b=*/false);
        }

        // ---- Epilogue: bias + store -------------------------------------
        // D layout: VGPR v -> (M = v + 8*khalf), lane column = lane&15.
        const float bv = b_lds[j * 16 + m_lane];
        float* obase = out
            + ((size_t)(r0 + khalf * 8) * N_FEAT + n) * D_OUT
            + j * 16 + m_lane;
        #pragma unroll
        for (int v = 0; v < 8; ++v) {
            obase[(size_t)v * row_stride] = acc[v] + bv;
        }
    }
}

extern "C" void kernel_launch(void* const* d_in, const int* in_sizes, int n_in,
                              void* d_out, int out_size, void* d_ws, size_t ws_size,
                              hipStream_t stream) {
    const float* x  = (const float*)d_in[0];   // [8192, 256, 64]
    const float* W  = (const float*)d_in[1];   // [256, 128, 64]
    const float* b  = (const float*)d_in[2];   // [256, 128]
    float* out      = (float*)d_out;           // [8192, 256, 128]

    dim3 grid(N_FEAT, B_ROWS / BM);            // (256, 64)
    dim3 block(256);                           // 8 wave32s
    nlinear_wmma_f32_kernel<<<grid, block, 0, stream>>>(x, W, b, out);
}